// ProdLayer_15942918602882
// MI455X (gfx1250) — compile-verified
//
#include <hip/hip_runtime.h>
#include <hip/hip_bf16.h>

// ProdLayer forward: element_mars[nids[p], :] = sum_c node_mars[cids[p,c], :]
// Shapes: N=P=500000, C=4, B=128, E=P+1. Pure bandwidth-bound gather/scatter:
// ~1.3 GB moved -> ~25-55us floor at 23.3 TB/s HBM (with 192MB-L2 reuse of the
// 256MB node_mars gather set). wave32: one wave handles one 512B output row
// (32 lanes x b128). CDNA5 path: async global->LDS staging (ASYNCcnt) + NT
// stores to keep the streaming output out of L2.

typedef float v4f __attribute__((ext_vector_type(4)));
typedef int   v4i __attribute__((ext_vector_type(4)));

#define WAVES_PER_BLOCK 8
#define BDIM (WAVES_PER_BLOCK * 32)

#define AS_GLOBAL __attribute__((address_space(1)))
#define AS_LOCAL  __attribute__((address_space(3)))

#define HAVE_ASYNC_LDS (__has_builtin(__builtin_amdgcn_global_load_async_to_lds_b128))

#if HAVE_ASYNC_LDS
// Builtin signature (from clang diagnostic): param0 = int4 in device/global AS,
// non-const. Cast through const_cast then an explicit address-space cast.
__device__ __forceinline__ void async_row16B_to_lds(const float* gsrc, float* ldst) {
    __builtin_amdgcn_global_load_async_to_lds_b128(
        (AS_GLOBAL v4i*)const_cast<float*>(gsrc),
        (AS_LOCAL  v4i*)ldst,
        0, 0);
}
#endif

__global__ __launch_bounds__(BDIM) void prod_fwd_b128_c4(
    const float* __restrict__ node_mars,
    const int*   __restrict__ nids,
    const int*   __restrict__ cids,
    float*       __restrict__ out,
    int P)
{
#if HAVE_ASYNC_LDS
    // 4 child-row staging slots (512B each) per wave; 8 waves -> 16KB of the
    // 320KB/WGP LDS. No cross-wave sharing -> no barriers needed, only
    // s_wait_asynccnt for same-wave completion.
    __shared__ float stage[WAVES_PER_BLOCK][4][128];
#endif
    const int wid  = (int)(threadIdx.x >> 5);   // wave in block (wave32)
    const int lane = (int)(threadIdx.x & 31);
    const int p    = (int)blockIdx.x * WAVES_PER_BLOCK + wid;  // one row per wave
    if (p >= P) return;

    // Wave-uniform index fetch: all 32 lanes hit the same 16B/4B lines
    // (single request, hardware broadcast).
    const int4 c4  = ((const int4*)cids)[p];
    const int  nid = nids[p];

    const int col = lane * 4;                    // 4 floats (16B) per lane; 32*16B = 512B row

    const float* g0 = node_mars + (size_t)c4.x * 128 + col;
    const float* g1 = node_mars + (size_t)c4.y * 128 + col;
    const float* g2 = node_mars + (size_t)c4.z * 128 + col;
    const float* g3 = node_mars + (size_t)c4.w * 128 + col;

    v4f a, b, c, d;

#if HAVE_ASYNC_LDS
    // CDNA5 async copy engine: global -> LDS, tracked by ASYNCcnt.
    async_row16B_to_lds(g0, &stage[wid][0][col]);
    async_row16B_to_lds(g1, &stage[wid][1][col]);
    async_row16B_to_lds(g2, &stage[wid][2][col]);
    async_row16B_to_lds(g3, &stage[wid][3][col]);
#if __has_builtin(__builtin_amdgcn_s_wait_asynccnt)
    __builtin_amdgcn_s_wait_asynccnt(0);
#else
    asm volatile("s_wait_asynccnt 0" ::: "memory");
#endif
    a = *(const v4f*)&stage[wid][0][col];        // ds_load_b128
    b = *(const v4f*)&stage[wid][1][col];
    c = *(const v4f*)&stage[wid][2][col];
    d = *(const v4f*)&stage[wid][3][col];
#else
    // Fallback: direct coalesced b128 gathers (default RT caching -> L2 reuse).
    a = *(const v4f*)g0;
    b = *(const v4f*)g1;
    c = *(const v4f*)g2;
    d = *(const v4f*)g3;
#endif

    v4f r = (a + b) + (c + d);

    // Write-once 256MB output stream: non-temporal store keeps it from
    // evicting the node_mars gather set out of the 192MB L2.
    __builtin_nontemporal_store(r, (v4f*)(out + (size_t)nid * 128 + col));
}

// Generic fallback for unexpected (B, C).
__global__ void prod_fwd_generic(
    const float* __restrict__ node_mars,
    const int*   __restrict__ nids,
    const int*   __restrict__ cids,
    float*       __restrict__ out,
    int P, int C, int B)
{
    long long t = (long long)blockIdx.x * blockDim.x + threadIdx.x;
    long long total = (long long)P * B;
    if (t >= total) return;
    int p = (int)(t / B);
    int b = (int)(t % B);
    float s = 0.0f;
    for (int c = 0; c < C; ++c)
        s += node_mars[(size_t)cids[(size_t)p * C + c] * B + b];
    __builtin_nontemporal_store(s, out + (size_t)nids[p] * B + b);
}

// Row 0 of element_mars is the untouched dummy row: passthrough from input.
__global__ void copy_row0(const float* __restrict__ element_mars,
                          float* __restrict__ out, int B)
{
    int i = (int)(blockIdx.x * blockDim.x + threadIdx.x);
    if (i < B) out[i] = element_mars[i];
}

extern "C" void kernel_launch(void* const* d_in, const int* in_sizes, int n_in,
                              void* d_out, int out_size, void* d_ws, size_t ws_size,
                              hipStream_t stream) {
    const float* node_mars    = (const float*)d_in[0];
    const float* element_mars = (const float*)d_in[1];
    const int*   nids         = (const int*)d_in[2];
    const int*   cids         = (const int*)d_in[3];
    float*       out          = (float*)d_out;

    const int P = in_sizes[2];                 // 500000
    const int C = in_sizes[3] / P;             // 4
    const int E = P + 1;
    const int B = in_sizes[1] / E;             // 128

    if (B == 128 && C == 4) {
        const int blocks = (P + WAVES_PER_BLOCK - 1) / WAVES_PER_BLOCK;  // 62500
        prod_fwd_b128_c4<<<blocks, BDIM, 0, stream>>>(node_mars, nids, cids, out, P);
    } else {
        const long long total = (long long)P * B;
        const int blocks = (int)((total + 255) / 256);
        prod_fwd_generic<<<blocks, 256, 0, stream>>>(node_mars, nids, cids, out, P, C, B);
    }
    copy_row0<<<(B + 127) / 128, 128, 0, stream>>>(element_mars, out, B);
}